// _DenseLayer_27066883899809
// MI455X (gfx1250) — compile-verified
//
#include <hip/hip_runtime.h>

// Problem constants (also derived from in_sizes at launch).
#define F_IN   128
#define F_HALF 64
#define GROWTH 64
#define BN_EPS 1e-5f

typedef __attribute__((ext_vector_type(2))) float v2f;
typedef __attribute__((ext_vector_type(8))) float v8f;

// ---------------------------------------------------------------------------
// K0: zero BN stat accumulators (256 floats) and both degree arrays.
__global__ void k_init(float* __restrict__ sums, float* __restrict__ degO,
                       float* __restrict__ degI, int n) {
    int i = blockIdx.x * blockDim.x + threadIdx.x;
    if (i < 256) sums[i] = 0.f;
    if (i < n) { degO[i] = 0.f; degI[i] = 0.f; }
}

// K1: d_out[n][j] = b[j]  (bias folded into output init; scatter adds on top)
__global__ void k_init_out(float* __restrict__ out, const float* __restrict__ b,
                           int total) {
    int i = blockIdx.x * blockDim.x + threadIdx.x;
    if (i < total) out[i] = b[i & (GROWTH - 1)];
}

// K2: per-column sum / sumsq of x = [x1 | x2].  128 threads = 1 column each,
// rows strided over blocks; one atomicAdd pair per (block, column).
__global__ void k_col_stats(const float* __restrict__ x1, const float* __restrict__ x2,
                            float* __restrict__ sum, float* __restrict__ sumsq, int n) {
    int c = threadIdx.x;  // 0..127
    const float* base = (c < F_HALF) ? (x1 + c) : (x2 + (c - F_HALF));
    float s = 0.f, s2 = 0.f;
    for (int r = blockIdx.x; r < n; r += gridDim.x) {
        float v = base[(long)r * F_HALF];
        s += v; s2 += v * v;
    }
    atomicAdd(&sum[c], s);
    atomicAdd(&sumsq[c], s2);
}

// K3: unweighted degree counts.
__global__ void k_degrees(const int* __restrict__ src, const int* __restrict__ dst,
                          float* __restrict__ degO, float* __restrict__ degI, int e) {
    int i = blockIdx.x * blockDim.x + threadIdx.x;
    if (i < e) {
        atomicAdd(&degO[src[i]], 1.f);
        atomicAdd(&degI[dst[i]], 1.f);
    }
}

// K4: BN scale/shift from batch stats; deg -> rsqrt(max(deg,1)) in place.
__global__ void k_finalize(const float* __restrict__ sum, const float* __restrict__ sumsq,
                           const float* __restrict__ gamma, const float* __restrict__ beta,
                           float* __restrict__ scale, float* __restrict__ shift,
                           float* __restrict__ degO, float* __restrict__ degI, int n) {
    int i = blockIdx.x * blockDim.x + threadIdx.x;
    if (i < F_IN) {
        float inv_n = 1.f / (float)n;
        float m   = sum[i] * inv_n;
        float var = sumsq[i] * inv_n - m * m;    // biased variance (jnp.var)
        float is  = rsqrtf(var + BN_EPS);
        float sc  = gamma[i] * is;
        scale[i] = sc;
        shift[i] = beta[i] - m * sc;
    }
    if (i < n) {
        degO[i] = rsqrtf(fmaxf(degO[i], 1.f));   // becomes norm_out
        degI[i] = rsqrtf(fmaxf(degI[i], 1.f));   // becomes norm_in
    }
}

// ---------------------------------------------------------------------------
// K5: fused BN + ReLU + (h @ W) * norm_out using V_WMMA_F32_16X16X4_F32.
// Block = 256 threads = 8 waves; block tile = 128 rows x K=128; each wave owns
// a 16-row strip and 4 accumulator tiles covering the 64 output columns.
// LDS: activation tile 128x128 (pad 129) + full W 128x64 (pad 65) ~= 97 KB.
#define TILE_M 128
#define LDH 129
#define LDW 65

__global__ __launch_bounds__(256) void k_bn_gemm(
        const float* __restrict__ x1, const float* __restrict__ x2,
        const float* __restrict__ Wm,
        const float* __restrict__ scale, const float* __restrict__ shift,
        const float* __restrict__ normO,
        float* __restrict__ hproj, int n) {
    __shared__ float sh[TILE_M * LDH];  // BN+ReLU'd activations
    __shared__ float sw[F_IN * LDW];    // weights

    const int t = threadIdx.x;
    const int rbase = blockIdx.x * TILE_M;

    // Cooperative coalesced load of the activation tile, BN+ReLU on the fly.
    // Tail rows are clamped (values masked at store) to keep EXEC uniform.
    for (int idx = t; idx < TILE_M * F_IN; idx += 256) {
        int row = idx >> 7, col = idx & (F_IN - 1);
        int rg = rbase + row; if (rg >= n) rg = n - 1;
        float v = (col < F_HALF) ? x1[(long)rg * F_HALF + col]
                                 : x2[(long)rg * F_HALF + (col - F_HALF)];
        v = v * scale[col] + shift[col];
        sh[row * LDH + col] = fmaxf(v, 0.f);
    }
    // Load W [128 x 64].
    for (int idx = t; idx < F_IN * GROWTH; idx += 256) {
        int row = idx >> 6, col = idx & (GROWTH - 1);
        sw[row * LDW + col] = Wm[idx];
    }
    __syncthreads();

    const int lane = t & 31;
    const int wave = t >> 5;       // 0..7 -> 16-row strip
    const int half = lane >> 4;    // 0|1 : K-offset +0 / +2 (A,B) ; M +0 / +8 (C,D)
    const int m    = lane & 15;    // A: row within strip ; B/C/D: column within tile
    const int mrow = wave * 16 + m;

    v8f acc0 = {}, acc1 = {}, acc2 = {}, acc3 = {};
    for (int ks = 0; ks < F_IN; ks += 4) {
        const int k = ks + half * 2;
        // A 16x4 fp32 fragment: v0 = K=k (lanes 0-15) / K=k+2 (lanes 16-31); v1 = +1.
        v2f a;  a.x  = sh[mrow * LDH + k];        a.y  = sh[mrow * LDH + k + 1];
        // B 4x16 fp32 fragments, one per 16-wide output column tile.
        v2f b0; b0.x = sw[k * LDW + m];           b0.y = sw[(k + 1) * LDW + m];
        v2f b1; b1.x = sw[k * LDW + 16 + m];      b1.y = sw[(k + 1) * LDW + 16 + m];
        v2f b2; b2.x = sw[k * LDW + 32 + m];      b2.y = sw[(k + 1) * LDW + 32 + m];
        v2f b3; b3.x = sw[k * LDW + 48 + m];      b3.y = sw[(k + 1) * LDW + 48 + m];
        acc0 = __builtin_amdgcn_wmma_f32_16x16x4_f32(false, a, false, b0, (short)0, acc0, false, false);
        acc1 = __builtin_amdgcn_wmma_f32_16x16x4_f32(false, a, false, b1, (short)0, acc1, false, false);
        acc2 = __builtin_amdgcn_wmma_f32_16x16x4_f32(false, a, false, b2, (short)0, acc2, false, false);
        acc3 = __builtin_amdgcn_wmma_f32_16x16x4_f32(false, a, false, b3, (short)0, acc3, false, false);
    }

    // C/D layout: VGPR v -> row = v + half*8, col = m. Stores are 16-lane
    // contiguous runs; scale by norm_out here.
    const int rowbase = rbase + wave * 16 + half * 8;
    for (int v = 0; v < 8; ++v) {
        int row = rowbase + v;
        if (row < n) {
            float s = normO[row];
            float* o = hproj + (long)row * GROWTH + m;
            o[0]  = acc0[v] * s;
            o[16] = acc1[v] * s;
            o[32] = acc2[v] * s;
            o[48] = acc3[v] * s;
        }
    }
}

// ---------------------------------------------------------------------------
// K6: edge scatter. One thread per (edge, feature). norm_in folded into the
// per-edge scale so this is the only pass over the edge feature traffic.
__global__ void k_scatter(const int* __restrict__ src, const int* __restrict__ dst,
                          const float* __restrict__ ew, const float* __restrict__ hproj,
                          const float* __restrict__ normI, float* __restrict__ out,
                          int e) {
    long idx = (long)blockIdx.x * blockDim.x + threadIdx.x;
    if (idx >= (long)e * GROWTH) return;
    int eid = (int)(idx >> 6);
    int j   = (int)(idx & (GROWTH - 1));
    int d = dst[eid];
    int s = src[eid];
    float w = ew[eid] * normI[d];
    atomicAdd(&out[(long)d * GROWTH + j], hproj[(long)s * GROWTH + j] * w);
}

// ---------------------------------------------------------------------------
extern "C" void kernel_launch(void* const* d_in, const int* in_sizes, int n_in,
                              void* d_out, int out_size, void* d_ws, size_t ws_size,
                              hipStream_t stream) {
    const float* x1    = (const float*)d_in[0];
    const float* x2    = (const float*)d_in[1];
    const int*   src   = (const int*)  d_in[2];
    const int*   dst   = (const int*)  d_in[3];
    const float* ew    = (const float*)d_in[4];
    const float* gamma = (const float*)d_in[5];
    const float* beta  = (const float*)d_in[6];
    const float* Wm    = (const float*)d_in[7];
    const float* b     = (const float*)d_in[8];
    float* out = (float*)d_out;

    const int n = in_sizes[0] / F_HALF;   // 50000
    const int e = in_sizes[2];            // 600000

    // Workspace layout (floats): stats[512] | norm_out[n] | norm_in[n] | hproj[n*64]
    float* ws    = (float*)d_ws;
    float* sum   = ws;
    float* sumsq = ws + 128;
    float* scale = ws + 256;
    float* shift = ws + 384;
    float* degO  = ws + 512;
    float* degI  = degO + n;
    float* hproj = degI + n;

    const int thr = 256;
    k_init    <<<(n + thr - 1) / thr, thr, 0, stream>>>(sum, degO, degI, n);
    const int total = n * GROWTH;
    k_init_out<<<(total + thr - 1) / thr, thr, 0, stream>>>(out, b, total);
    k_col_stats<<<256, 128, 0, stream>>>(x1, x2, sum, sumsq, n);
    k_degrees <<<(e + thr - 1) / thr, thr, 0, stream>>>(src, dst, degO, degI, e);
    k_finalize<<<(n + thr - 1) / thr, thr, 0, stream>>>(sum, sumsq, gamma, beta,
                                                        scale, shift, degO, degI, n);
    k_bn_gemm <<<(n + TILE_M - 1) / TILE_M, 256, 0, stream>>>(x1, x2, Wm, scale, shift,
                                                              degO, hproj, n);
    const long etotal = (long)e * GROWTH;
    k_scatter <<<(int)((etotal + thr - 1) / thr), thr, 0, stream>>>(src, dst, ew, hproj,
                                                                    degI, out, e);
}